// Encoder_22359599743561
// MI455X (gfx1250) — compile-verified
//
#include <hip/hip_runtime.h>

// ---------------------------------------------------------------------------
// VGAE encoder (2x GCNConv) for gfx1250 (MI455X).
//   h  = relu(norm-propagate(x @ W1) + b1)
//   mu = norm-propagate(h @ Wmu) + bmu ; logvar likewise with Wlv/blv
// Dense GEMMs use V_WMMA_F32_16X16X4_F32 (exact fp32, matches reference).
// Graph propagation uses global_atomic_add_f32 scatter (memory-bound part:
// ~2.7 GB total traffic -> ~115us at 23.3 TB/s; hot buffers fit in 192MB L2).
// ---------------------------------------------------------------------------

typedef float v2f __attribute__((ext_vector_type(2)));
typedef float v8f __attribute__((ext_vector_type(8)));

// ---------------- degree / norm ----------------

__global__ void k_fill1(float* __restrict__ p, int n) {
    int i = blockIdx.x * blockDim.x + threadIdx.x;
    if (i < n) p[i] = 1.0f;               // self-loop contributes 1 to degree
}

__global__ void k_deg_count(const long long* __restrict__ dst,
                            float* __restrict__ deg, int E) {
    int e = blockIdx.x * blockDim.x + threadIdx.x;
    if (e < E) unsafeAtomicAdd(&deg[(size_t)dst[e]], 1.0f);
}

__global__ void k_rsqrt_inplace(float* __restrict__ p, int n) {
    int i = blockIdx.x * blockDim.x + threadIdx.x;
    if (i < n) p[i] = rsqrtf(p[i]);       // deg >= 1 always (self loop)
}

// ---------------- pack W2 = [Wmu | Wlv], b2 = [bmu | blv] ----------------

__global__ void k_pack_w2(const float* __restrict__ Wmu, const float* __restrict__ Wlv,
                          const float* __restrict__ bmu, const float* __restrict__ blv,
                          float* __restrict__ W2, float* __restrict__ b2,
                          int hid, int cout) {
    int i = blockIdx.x * blockDim.x + threadIdx.x;
    int N2 = 2 * cout;
    int total = hid * N2;
    if (i < total) {
        int k = i / N2, j = i % N2;
        W2[i] = (j < cout) ? Wmu[k * cout + j] : Wlv[k * cout + (j - cout)];
    } else if (i < total + N2) {
        int j = i - total;
        b2[j] = (j < cout) ? bmu[j] : blv[j - cout];
    }
}

// ---------------- WMMA fp32 GEMM: C[M,N] = A[M,K] @ W[K,N] ----------------
// N must be 128 (8 waves x one 16x16 tile each). K multiple of 4, K <= 256.
// 16-row A strip staged in LDS with +4-float row pad (bank-conflict free).
// ISA frag layouts (wave32):
//   A 16x4 f32 : lane m=lane&15; VGPR0=A[m][kb], VGPR1=A[m][kb+1], kb=(lane>=16)?2:0
//   B 4x16 f32 : lane n=lane&15; VGPR0=B[kb][n], VGPR1=B[kb+1][n]
//   C/D 16x16  : VGPR r -> row r (lanes 0-15) / row r+8 (lanes 16-31)

#define GEMM_LDA 260   // 256 max K + 4 pad

__global__ void __launch_bounds__(256)
k_gemm_wmma(const float* __restrict__ A, const float* __restrict__ W,
            float* __restrict__ C, int M, int K, int N) {
    __shared__ float sA[16 * GEMM_LDA];
    const int tid  = threadIdx.x;
    const int row0 = blockIdx.x * 16;

    // cooperative, coalesced B128 loads of the 16 x K A-strip
    for (int i = tid * 4; i < 16 * K; i += blockDim.x * 4) {
        int r = i / K, c = i % K;
        int gr = row0 + r;
        float4 v = make_float4(0.f, 0.f, 0.f, 0.f);
        if (gr < M) v = *reinterpret_cast<const float4*>(A + (size_t)gr * K + c);
        *reinterpret_cast<float4*>(&sA[r * GEMM_LDA + c]) = v;
    }
    __syncthreads();

    const int wave = tid >> 5;
    const int lane = tid & 31;
    const int m    = lane & 15;
    const int col  = wave * 16 + m;       // global N column (N == 128)
    const int hi   = lane >> 4;
    const int kb   = hi * 2;

    v8f acc0 = {};
    v8f acc1 = {};
    int k0 = 0;
    for (; k0 + 8 <= K; k0 += 8) {
        {
            v2f a, b;
            a.x = sA[m * GEMM_LDA + k0 + kb];
            a.y = sA[m * GEMM_LDA + k0 + kb + 1];
            b.x = W[(size_t)(k0 + kb) * N + col];
            b.y = W[(size_t)(k0 + kb + 1) * N + col];
            acc0 = __builtin_amdgcn_wmma_f32_16x16x4_f32(false, a, false, b,
                                                         (short)0, acc0, false, false);
        }
        {
            v2f a, b;
            const int kk = k0 + 4;
            a.x = sA[m * GEMM_LDA + kk + kb];
            a.y = sA[m * GEMM_LDA + kk + kb + 1];
            b.x = W[(size_t)(kk + kb) * N + col];
            b.y = W[(size_t)(kk + kb + 1) * N + col];
            acc1 = __builtin_amdgcn_wmma_f32_16x16x4_f32(false, a, false, b,
                                                         (short)0, acc1, false, false);
        }
    }
    for (; k0 < K; k0 += 4) {
        v2f a, b;
        a.x = sA[m * GEMM_LDA + k0 + kb];
        a.y = sA[m * GEMM_LDA + k0 + kb + 1];
        b.x = W[(size_t)(k0 + kb) * N + col];
        b.y = W[(size_t)(k0 + kb + 1) * N + col];
        acc0 = __builtin_amdgcn_wmma_f32_16x16x4_f32(false, a, false, b,
                                                     (short)0, acc0, false, false);
    }

    const int rbase = hi * 8;
    for (int r = 0; r < 8; ++r) {
        int gr = row0 + rbase + r;
        if (gr < M) C[(size_t)gr * N + col] = acc0[r] + acc1[r];
    }
}

// ---------------- propagation ----------------
// Split-destination addressing: col c < split -> pA[v*split + c]
//                               else          -> pB[v*(C-split) + (c-split)]

__global__ void k_prop_init(const float* __restrict__ H, const float* __restrict__ dinv,
                            const float* __restrict__ bias, int n, int C,
                            float* __restrict__ pA, float* __restrict__ pB, int split) {
    size_t idx = (size_t)blockIdx.x * blockDim.x + threadIdx.x;
    if (idx >= (size_t)n * C) return;
    int v = (int)(idx / C);
    int c = (int)(idx % C);
    float di  = dinv[v];
    float val = di * di * H[idx] + bias[c];
    if (c < split) pA[(size_t)v * split + c] = val;
    else           pB[(size_t)v * (C - split) + (c - split)] = val;
}

// One wave per edge: B128 gather of H[src], 4 fp32 global atomics per lane.
__global__ void __launch_bounds__(256)
k_scatter(const float* __restrict__ H, const float* __restrict__ dinv,
          const long long* __restrict__ src, const long long* __restrict__ dst,
          int E, int C, float* __restrict__ pA, float* __restrict__ pB, int split) {
    int e = blockIdx.x * 8 + (threadIdx.x >> 5);
    if (e >= E) return;
    const int lane = threadIdx.x & 31;
    const size_t s = (size_t)src[e];
    const size_t d = (size_t)dst[e];
    const float norm = dinv[s] * dinv[d];
    const float* hs = H + s * C;

    for (int c0 = lane * 4; c0 < C; c0 += 128) {
        const float4 hv = *reinterpret_cast<const float4*>(hs + c0);
        float* out = (c0 < split) ? (pA + d * split + c0)
                                  : (pB + d * (size_t)(C - split) + (c0 - split));
        unsafeAtomicAdd(out + 0, hv.x * norm);
        unsafeAtomicAdd(out + 1, hv.y * norm);
        unsafeAtomicAdd(out + 2, hv.z * norm);
        unsafeAtomicAdd(out + 3, hv.w * norm);
    }
}

__global__ void k_relu(float* __restrict__ p, size_t total) {
    size_t i = (size_t)blockIdx.x * blockDim.x + threadIdx.x;
    if (i < total) p[i] = fmaxf(p[i], 0.0f);
}

// ---------------- host orchestration ----------------

extern "C" void kernel_launch(void* const* d_in, const int* in_sizes, int n_in,
                              void* d_out, int out_size, void* d_ws, size_t ws_size,
                              hipStream_t stream) {
    const float*     x    = (const float*)d_in[0];
    const long long* edge = (const long long*)d_in[1];   // int64 [2, E]
    const float*     W1   = (const float*)d_in[2];
    const float*     b1   = (const float*)d_in[3];
    const float*     Wmu  = (const float*)d_in[4];
    const float*     bmu  = (const float*)d_in[5];
    const float*     Wlv  = (const float*)d_in[6];
    const float*     blv  = (const float*)d_in[7];

    const int hid  = in_sizes[3];            // 128
    const int cout = in_sizes[5];            // 64
    const int cin  = in_sizes[2] / hid;      // 256
    const int n    = in_sizes[0] / cin;      // 100000
    const int E    = in_sizes[1] / 2;        // 1600000
    const int N2   = 2 * cout;               // 128 (== hid here)

    const long long* src = edge;
    const long long* dst = edge + E;

    // workspace carve-up (fp32), 256B-aligned chunks; total ~103 MB
    float* ws  = (float*)d_ws;
    size_t off = 0;
    auto alloc = [&](size_t cnt) {
        float* p = ws + off;
        off += (cnt + 63) & ~(size_t)63;
        return p;
    };
    const size_t Hcols = (size_t)((hid > N2) ? hid : N2);
    float* dinv = alloc(n);                      // degree, then rsqrt(deg)
    float* W2   = alloc((size_t)hid * N2);
    float* b2   = alloc(N2);
    float* H    = alloc((size_t)n * Hcols);      // GEMM output (both layers)
    float* h    = alloc((size_t)n * hid);        // layer-1 activations

    float* mu = (float*)d_out;
    float* lv = mu + (size_t)n * cout;

    const int T = 256;
    // degrees and symmetric norm
    k_fill1<<<(n + T - 1) / T, T, 0, stream>>>(dinv, n);
    k_deg_count<<<(E + T - 1) / T, T, 0, stream>>>(dst, dinv, E);
    k_rsqrt_inplace<<<(n + T - 1) / T, T, 0, stream>>>(dinv, n);

    // pack layer-2 weights/bias
    k_pack_w2<<<(hid * N2 + N2 + T - 1) / T, T, 0, stream>>>(Wmu, Wlv, bmu, blv,
                                                             W2, b2, hid, cout);

    // ---- layer 1: H = x @ W1 ; h = relu(propagate(H) + b1) ----
    k_gemm_wmma<<<(n + 15) / 16, T, 0, stream>>>(x, W1, H, n, cin, hid);
    k_prop_init<<<(int)(((size_t)n * hid + T - 1) / T), T, 0, stream>>>(
        H, dinv, b1, n, hid, h, h, hid);
    k_scatter<<<(E + 7) / 8, T, 0, stream>>>(H, dinv, src, dst, E, hid, h, h, hid);
    k_relu<<<(int)(((size_t)n * hid + T - 1) / T), T, 0, stream>>>(h, (size_t)n * hid);

    // ---- layer 2: H = h @ [Wmu|Wlv] ; mu/logvar = propagate(H) + b2 ----
    k_gemm_wmma<<<(n + 15) / 16, T, 0, stream>>>(h, W2, H, n, hid, N2);
    k_prop_init<<<(int)(((size_t)n * N2 + T - 1) / T), T, 0, stream>>>(
        H, dinv, b2, n, N2, mu, lv, cout);
    k_scatter<<<(E + 7) / 8, T, 0, stream>>>(H, dinv, src, dst, E, N2, mu, lv, cout);
}